// myModel_82643760710209
// MI455X (gfx1250) — compile-verified
//
#include <hip/hip_runtime.h>
#include <hip/hip_bf16.h>

typedef __attribute__((ext_vector_type(2))) float v2f;
typedef __attribute__((ext_vector_type(8))) float v8f;

#define D 128

// ---------------------------------------------------------------------------
// deg[i] = 1.0f  (self-loop)
__global__ void k_init_deg(float* __restrict__ deg, int n) {
    int i = blockIdx.x * blockDim.x + threadIdx.x;
    if (i < n) deg[i] = 1.0f;
}

// deg[dst[e]] += 1
__global__ void k_accum_deg(const int* __restrict__ dst, float* __restrict__ deg, int e) {
    int i = blockIdx.x * blockDim.x + threadIdx.x;
    if (i < e) atomicAdd(&deg[dst[i]], 1.0f);
}

// dinv[i] = rsqrt(deg[i])   (deg >= 1 always due to self-loop)
__global__ void k_rsqrt(float* __restrict__ deg, int n) {
    int i = blockIdx.x * blockDim.x + threadIdx.x;
    if (i < n) deg[i] = __frsqrt_rn(deg[i]);
}

// ---------------------------------------------------------------------------
// H[nrows x 128] = X[nrows x 128] @ W[128 x 128], full f32 via WMMA 16x16x4.
// One wave32 computes one 16x16 tile of H, looping K in steps of 4.
__global__ void k_gemm_wmma_f32(const float* __restrict__ X,
                                const float* __restrict__ W,
                                float* __restrict__ H, int nrows) {
    const int lane  = threadIdx.x & 31;
    const int wave  = (blockIdx.x * blockDim.x + threadIdx.x) >> 5;
    const int colTiles = D / 16;                 // 8
    const int rowTile  = wave / colTiles;
    const int colTile  = wave % colTiles;
    if (rowTile * 16 >= nrows) return;

    const int m    = lane & 15;                  // row within A tile / col within B tile
    const int half = lane >> 4;                  // 0: K pair {k,k+1}; 1: {k+2,k+3}
    const long long row = (long long)rowTile * 16 + m;
    const int col  = colTile * 16 + m;           // N index for B and C/D

    v8f c = {};
    const float* xrow = X + row * D;
    #pragma unroll 4
    for (int k = 0; k < D; k += 4) {
        const int kb = k + half * 2;
        v2f a, b;
        a.x = xrow[kb];
        a.y = xrow[kb + 1];
        b.x = W[kb * D + col];
        b.y = W[(kb + 1) * D + col];
        // D = A(16x4) * B(4x16) + C   (f32, round-to-nearest-even)
        c = __builtin_amdgcn_wmma_f32_16x16x4_f32(
                /*neg_a=*/false, a, /*neg_b=*/false, b,
                /*c_mod=*/(short)0, c, /*reuse_a=*/false, /*reuse_b=*/false);
    }

    // C/D layout: VGPR r -> M = r (lanes 0-15) or M = r+8 (lanes 16-31), N = lane&15
    #pragma unroll
    for (int r = 0; r < 8; ++r) {
        const long long mr = (long long)rowTile * 16 + r + 8 * half;
        H[mr * D + col] = c[r];
    }
}

// ---------------------------------------------------------------------------
// acc[i][d] = dinv[i]^2 * H[i][d]     (self-loop term; initializes accumulator)
__global__ void k_selfloop_init(float* __restrict__ acc,
                                const float* __restrict__ H,
                                const float* __restrict__ dinv, int n) {
    long long gid = (long long)blockIdx.x * blockDim.x + threadIdx.x;
    long long total = (long long)n * D;
    if (gid >= total) return;
    int node = (int)(gid >> 7);                  // D == 128
    float w = dinv[node];
    acc[gid] = w * w * H[gid];
}

// acc[dst][:] += dinv[src]*dinv[dst] * H[src][:]   (32 lanes/edge, float4 each)
__global__ void k_scatter_edges(const int* __restrict__ src,
                                const int* __restrict__ dst,
                                const float* __restrict__ dinv,
                                const float* __restrict__ H,
                                float* __restrict__ acc, int e) {
    long long gid = (long long)blockIdx.x * blockDim.x + threadIdx.x;
    int edge = (int)(gid >> 5);
    if (edge >= e) return;
    int q = (int)(gid & 31);                     // which float4 of the 128-f row
    int s = src[edge];
    int d = dst[edge];
    float norm = dinv[s] * dinv[d];
    const float4 hv = ((const float4*)(H + (long long)s * D))[q];
    float* out = acc + (long long)d * D + q * 4;
    atomicAdd(out + 0, norm * hv.x);
    atomicAdd(out + 1, norm * hv.y);
    atomicAdd(out + 2, norm * hv.z);
    atomicAdd(out + 3, norm * hv.w);
}

// out[i][d] = relu(acc[i][d] + b[d])    (out may alias acc)
__global__ void k_bias_relu(float* __restrict__ out,
                            const float* __restrict__ acc,
                            const float* __restrict__ b, int n) {
    long long gid = (long long)blockIdx.x * blockDim.x + threadIdx.x;
    long long total = (long long)n * D;
    if (gid >= total) return;
    float v = acc[gid] + b[(int)(gid & (D - 1))];
    out[gid] = v > 0.0f ? v : 0.0f;
}

// ---------------------------------------------------------------------------
extern "C" void kernel_launch(void* const* d_in, const int* in_sizes, int n_in,
                              void* d_out, int out_size, void* d_ws, size_t ws_size,
                              hipStream_t stream) {
    const float* x    = (const float*)d_in[0];
    const int*   ei   = (const int*)d_in[1];
    const float* W1   = (const float*)d_in[2];
    const float* b1   = (const float*)d_in[3];
    const float* W2   = (const float*)d_in[4];
    const float* b2   = (const float*)d_in[5];
    float* out        = (float*)d_out;

    const int n = in_sizes[0] / D;               // 50000
    const int e = in_sizes[1] / 2;               // 600000
    const int* src = ei;                         // edge_index[0]
    const int* dst = ei + e;                     // edge_index[1]

    // Workspace layout: dinv[n] | bufA[n*D] | bufB[n*D]
    float* dinv = (float*)d_ws;
    float* bufA = dinv + (((long long)n + 63) / 64) * 64;
    float* bufB = bufA + (long long)n * D;

    const int T = 256;
    const long long nd = (long long)n * D;
    dim3 gN((n + T - 1) / T);
    dim3 gE((e + T - 1) / T);
    dim3 gND((unsigned)((nd + T - 1) / T));
    const long long waves = ((long long)((n + 15) / 16)) * (D / 16);
    dim3 gGemm((unsigned)((waves * 32 + T - 1) / T));
    const long long escat = (long long)e * 32;
    dim3 gScat((unsigned)((escat + T - 1) / T));

    // Degrees (shared by both layers)
    k_init_deg<<<gN, T, 0, stream>>>(dinv, n);
    k_accum_deg<<<gE, T, 0, stream>>>(dst, dinv, e);
    k_rsqrt<<<gN, T, 0, stream>>>(dinv, n);

    // Layer 1: h1 = x@W1 -> bufA; acc1 -> bufB; x2 = relu(acc1+b1) -> bufA
    k_gemm_wmma_f32<<<gGemm, T, 0, stream>>>(x, W1, bufA, n);
    k_selfloop_init<<<gND, T, 0, stream>>>(bufB, bufA, dinv, n);
    k_scatter_edges<<<gScat, T, 0, stream>>>(src, dst, dinv, bufA, bufB, e);
    k_bias_relu<<<gND, T, 0, stream>>>(bufA, bufB, b1, n);

    // Layer 2: h2 = x2@W2 -> bufB; acc2 -> d_out; out = relu(acc2+b2) in-place
    k_gemm_wmma_f32<<<gGemm, T, 0, stream>>>(bufA, W2, bufB, n);
    k_selfloop_init<<<gND, T, 0, stream>>>(out, bufB, dinv, n);
    k_scatter_edges<<<gScat, T, 0, stream>>>(src, dst, dinv, bufB, out, e);
    k_bias_relu<<<gND, T, 0, stream>>>(out, out, b2, n);
}